// ReferenceTopKRouter_28767690949267
// MI455X (gfx1250) — compile-verified
//
#include <hip/hip_runtime.h>
#include <hip/hip_bf16.h>

typedef __attribute__((ext_vector_type(2))) float v2f;
typedef __attribute__((ext_vector_type(4))) float v4f;
typedef __attribute__((ext_vector_type(8))) float v8f;

#define WAVES_PER_BLOCK 8
#define TOKENS_PER_WAVE 16
#define LDS_STRIDE 65   // 64 experts + 1 pad -> conflict-free per-token column reads
#define NUM_EXPERTS 64
#define HDIM 4096       // compile-time H: B-tile offsets fold into 24-bit immediates
#define PF_DIST 2048    // prefetch distance in floats (8 KB ahead on the x stream)

// One wave32 computes a 16-token x 64-expert logit tile with fp32 WMMA
// (D = A[16x4] * B[4x16] + C, K-loop over H, 2-stage software pipeline),
// then does bias + top-2 + softmax + scatter for its 16 tokens.
// `hdim` is passed at runtime purely to keep the loop trip count unknown,
// so the compiler honors the partial-unroll pragma instead of flattening
// ~16K instructions of straight-line code that would blow the I-cache.
__global__ __launch_bounds__(256) void topk_router_kernel(
    const float* __restrict__ x,      // [N, H] row-major
    const float* __restrict__ w,      // [E=64, H] row-major
    const float* __restrict__ bias,   // [64]
    float* __restrict__ out,          // [N, 64] (zeros except top-2)
    int* __restrict__ idx_out,        // [N, 2]
    int hdim)                         // == HDIM at runtime
{
    const int lane = threadIdx.x & 31;
    const int wave = threadIdx.x >> 5;
    const int m    = lane & 15;       // row-within-tile (A) / expert-within-tile (B)
    const int half = lane >> 4;       // selects K pair {0,1} vs {2,3}
    const int T0   = (blockIdx.x * WAVES_PER_BLOCK + wave) * TOKENS_PER_WAVE;

    __shared__ float slds[WAVES_PER_BLOCK][TOKENS_PER_WAVE * LDS_STRIDE];

    // A feed: lane (m, half) reads x[T0+m][h + 2*half .. +1]
    const float* ap = x + (size_t)(T0 + m) * HDIM + 2 * half;
    // B feed: lane (m, half) reads w[etile*16+m][h + 2*half .. +1]
    const float* bp = w + (size_t)m * HDIM + 2 * half;

    v8f acc0 = {}, acc1 = {}, acc2 = {}, acc3 = {};

    // ---- software pipeline: 2 stages of (A, B0..B3) in flight ----
    v2f aA  = *(const v2f*)(ap);
    v2f b0A = *(const v2f*)(bp);
    v2f b1A = *(const v2f*)(bp + 16 * HDIM);
    v2f b2A = *(const v2f*)(bp + 32 * HDIM);
    v2f b3A = *(const v2f*)(bp + 48 * HDIM);

    v2f aB  = *(const v2f*)(ap + 4);
    v2f b0B = *(const v2f*)(bp + 4);
    v2f b1B = *(const v2f*)(bp + 4 + 16 * HDIM);
    v2f b2B = *(const v2f*)(bp + 4 + 32 * HDIM);
    v2f b3B = *(const v2f*)(bp + 4 + 48 * HDIM);

    #pragma unroll 2
    for (int h = 8; h < hdim; h += 8) {
        // issue loads for step h (consumed two halves later)
        v2f aN0  = *(const v2f*)(ap + h);
        v2f b0N0 = *(const v2f*)(bp + h);
        v2f b1N0 = *(const v2f*)(bp + h + 16 * HDIM);
        v2f b2N0 = *(const v2f*)(bp + h + 32 * HDIM);
        v2f b3N0 = *(const v2f*)(bp + h + 48 * HDIM);
        // WGP-scope prefetch: pull the x stream into all cache levels
        __builtin_prefetch(ap + h + PF_DIST, 0, 3);

        // consume stage A (loaded at h-8): only older loads must complete
        acc0 = __builtin_amdgcn_wmma_f32_16x16x4_f32(false, aA, false, b0A, (short)0, acc0, false, false);
        acc1 = __builtin_amdgcn_wmma_f32_16x16x4_f32(false, aA, false, b1A, (short)0, acc1, false, false);
        acc2 = __builtin_amdgcn_wmma_f32_16x16x4_f32(false, aA, false, b2A, (short)0, acc2, false, false);
        acc3 = __builtin_amdgcn_wmma_f32_16x16x4_f32(false, aA, false, b3A, (short)0, acc3, false, false);

        // issue loads for step h+4
        v2f aN1  = *(const v2f*)(ap + h + 4);
        v2f b0N1 = *(const v2f*)(bp + h + 4);
        v2f b1N1 = *(const v2f*)(bp + h + 4 + 16 * HDIM);
        v2f b2N1 = *(const v2f*)(bp + h + 4 + 32 * HDIM);
        v2f b3N1 = *(const v2f*)(bp + h + 4 + 48 * HDIM);

        // consume stage B (loaded at h-4)
        acc0 = __builtin_amdgcn_wmma_f32_16x16x4_f32(false, aB, false, b0B, (short)0, acc0, false, false);
        acc1 = __builtin_amdgcn_wmma_f32_16x16x4_f32(false, aB, false, b1B, (short)0, acc1, false, false);
        acc2 = __builtin_amdgcn_wmma_f32_16x16x4_f32(false, aB, false, b2B, (short)0, acc2, false, false);
        acc3 = __builtin_amdgcn_wmma_f32_16x16x4_f32(false, aB, false, b3B, (short)0, acc3, false, false);

        // rotate stages (renamed away by the x2 unroll)
        aA = aN0; b0A = b0N0; b1A = b1N0; b2A = b2N0; b3A = b3N0;
        aB = aN1; b0B = b0N1; b1B = b1N1; b2B = b2N1; b3B = b3N1;
    }

    // drain the pipeline
    acc0 = __builtin_amdgcn_wmma_f32_16x16x4_f32(false, aA, false, b0A, (short)0, acc0, false, false);
    acc1 = __builtin_amdgcn_wmma_f32_16x16x4_f32(false, aA, false, b1A, (short)0, acc1, false, false);
    acc2 = __builtin_amdgcn_wmma_f32_16x16x4_f32(false, aA, false, b2A, (short)0, acc2, false, false);
    acc3 = __builtin_amdgcn_wmma_f32_16x16x4_f32(false, aA, false, b3A, (short)0, acc3, false, false);
    acc0 = __builtin_amdgcn_wmma_f32_16x16x4_f32(false, aB, false, b0B, (short)0, acc0, false, false);
    acc1 = __builtin_amdgcn_wmma_f32_16x16x4_f32(false, aB, false, b1B, (short)0, acc1, false, false);
    acc2 = __builtin_amdgcn_wmma_f32_16x16x4_f32(false, aB, false, b2B, (short)0, acc2, false, false);
    acc3 = __builtin_amdgcn_wmma_f32_16x16x4_f32(false, aB, false, b3B, (short)0, acc3, false, false);

    // C/D layout: VGPR r of lane L = logit[token = r + 8*(L>>4)][expert = etile*16 + (L&15)]
    const float b0s = bias[ 0 + m];
    const float b1s = bias[16 + m];
    const float b2s = bias[32 + m];
    const float b3s = bias[48 + m];

    float* row = slds[wave];
    #pragma unroll
    for (int r = 0; r < 8; ++r) {
        const int tok = r + 8 * half;
        row[tok * LDS_STRIDE +  0 + m] = acc0[r] + b0s;
        row[tok * LDS_STRIDE + 16 + m] = acc1[r] + b1s;
        row[tok * LDS_STRIDE + 32 + m] = acc2[r] + b2s;
        row[tok * LDS_STRIDE + 48 + m] = acc3[r] + b3s;
    }

    __syncthreads();

    // Lanes 0..15: top-2 + softmax + scatter for token T0+lane.
    if (lane < 16) {
        const float* lrow = slds[wave] + lane * LDS_STRIDE;
        float best1 = -__builtin_inff(), best2 = -__builtin_inff();
        int i1 = 0, i2 = 0;
        #pragma unroll 8
        for (int e = 0; e < NUM_EXPERTS; ++e) {
            const float v = lrow[e];
            if (v > best1)      { best2 = best1; i2 = i1; best1 = v; i1 = e; }
            else if (v > best2) { best2 = v; i2 = e; }
        }
        // softmax over {best1, best2}
        const float e2 = __expf(best2 - best1);
        const float w1 = 1.0f / (1.0f + e2);
        const float w2 = 1.0f - w1;

        const int token = T0 + lane;
        float* orow = out + (size_t)token * NUM_EXPERTS;
        v4f zero = {};
        #pragma unroll
        for (int j = 0; j < NUM_EXPERTS / 4; ++j)
            ((v4f*)orow)[j] = zero;                 // b128 zero fill
        orow[i1] = w1;                              // same-lane stores stay ordered
        orow[i2] = w2;
        idx_out[token * 2 + 0] = i1;
        idx_out[token * 2 + 1] = i2;
    }
}

extern "C" void kernel_launch(void* const* d_in, const int* in_sizes, int n_in,
                              void* d_out, int out_size, void* d_ws, size_t ws_size,
                              hipStream_t stream) {
    const float* x    = (const float*)d_in[0];   // [B,S,H] fp32
    const float* w    = (const float*)d_in[1];   // [E,H]   fp32
    const float* bias = (const float*)d_in[2];   // [E]     fp32

    const int E = in_sizes[2];                   // 64
    const int H = in_sizes[1] / E;               // 4096 (kernel assumes HDIM)
    const int N = in_sizes[0] / H;               // 16384

    float* out = (float*)d_out;                  // [N,64] fp32 ...
    int* idx   = (int*)(out + (size_t)N * E);    // ... then [N,2] int32

    const int tokens_per_block = WAVES_PER_BLOCK * TOKENS_PER_WAVE; // 128
    const int blocks = N / tokens_per_block;                        // 128

    topk_router_kernel<<<blocks, 256, 0, stream>>>(x, w, bias, out, idx, H);
}